// ProximalGradient_43920335569175
// MI455X (gfx1250) — compile-verified
//
#include <hip/hip_runtime.h>
#include <math.h>

#define HH 512
#define WW 512
#define CC 64
#define WMM 128
#define LMBDA 0.1f

typedef __attribute__((ext_vector_type(2))) float v2f;
typedef __attribute__((ext_vector_type(8))) float v8f;
typedef __attribute__((ext_vector_type(4))) unsigned int u32x4;
typedef __attribute__((ext_vector_type(8))) unsigned int u32x8;

// ---------------- WMMA helper: D = A(16x4,f32) * B(4x16,f32) + C -------------
__device__ __forceinline__ v8f wmma_f32_k4(v2f a, v2f b, v8f c) {
  // 8 args: (neg_a, A, neg_b, B, c_mod, C, reuse_a, reuse_b)
  return __builtin_amdgcn_wmma_f32_16x16x4_f32(false, a, false, b, (short)0, c,
                                               false, false);
}

// ---------------- async copy to LDS (CDNA5, ASYNCcnt-tracked) ----------------
__device__ __forceinline__ void async_b32(unsigned int lds_addr,
                                          const void* gaddr) {
  asm volatile("global_load_async_to_lds_b32 %0, %1, off" ::"v"(lds_addr),
               "v"((unsigned long long)(uintptr_t)gaddr)
               : "memory");
}
__device__ __forceinline__ void wait_async0() {
  asm volatile("s_wait_asynccnt 0x0" ::: "memory");
}
__device__ __forceinline__ unsigned int lds_off(const void* p) {
  // generic LDS pointer: low 32 bits are the wave-relative LDS byte address
  return (unsigned int)(uintptr_t)p;
}

// ---------------- Tensor Data Mover: 2D tile load, TENSORcnt-tracked ---------
// D# per CDNA5 ISA. data_size = 4 bytes.
// Loads a (tile_d0 x tile_d1) f32 tile from a row-major tensor with row stride
// `stride0` elements into LDS, inserting `pad_amt+1` DWORDs of padding every
// 2^(pad_iv+1) DWORDs when pad_en=1.
__device__ __forceinline__ void tdm_load_2d_f32(
    unsigned int lds_addr, const void* gaddr, unsigned int tensor_d0,
    unsigned int tensor_d1, unsigned int tile_d0, unsigned int tile_d1,
    unsigned int stride0, unsigned int pad_en, unsigned int pad_iv,
    unsigned int pad_amt) {
  unsigned long long ga = (unsigned long long)(uintptr_t)gaddr;
  u32x4 g0;
  g0[0] = 1u;                                   // count=1 (valid user D#)
  g0[1] = lds_addr;                             // lds_addr[31:0]
  g0[2] = (unsigned int)ga;                     // global_addr[31:0]
  g0[3] = (unsigned int)((ga >> 32) & 0x01FFFFFFull) | (2u << 30);  // type=2
  u32x8 g1;
  g1[0] = (2u << 16) | (pad_en << 20) | (pad_iv << 22) | (pad_amt << 25);
  g1[1] = (tensor_d0 & 0xFFFFu) << 16;          // tensor_dim0[15:0]
  g1[2] = (tensor_d0 >> 16) | ((tensor_d1 & 0xFFFFu) << 16);
  g1[3] = (tensor_d1 >> 16) | (tile_d0 << 16);  // tile_dim0
  g1[4] = tile_d1;                              // tile_dim1 ; tile_dim2 = 0
  g1[5] = stride0;                              // tensor_dim0_stride[31:0]
  g1[6] = 0u;
  g1[7] = 0u;
  u32x4 gz = {0u, 0u, 0u, 0u};                  // dims/tiles 0 => unused
  asm volatile("tensor_load_to_lds %0, %1, %2, %3" ::"s"(g0), "s"(g1), "s"(gz),
               "s"(gz)
               : "memory");
}
__device__ __forceinline__ void wait_tensor0() {
  __builtin_amdgcn_s_wait_tensorcnt(0);
}

// ---------------- small float4 helpers ---------------------------------------
__device__ __forceinline__ float4 ld4(const float* p) { return *(const float4*)p; }
__device__ __forceinline__ void st4(float* p, float4 v) { *(float4*)p = v; }
__device__ __forceinline__ float4 f4zero() { return make_float4(0.f, 0.f, 0.f, 0.f); }
__device__ __forceinline__ float4 f4add(float4 a, float4 b) {
  return make_float4(a.x + b.x, a.y + b.y, a.z + b.z, a.w + b.w);
}
__device__ __forceinline__ float4 f4sub(float4 a, float4 b) {
  return make_float4(a.x - b.x, a.y - b.y, a.z - b.z, a.w - b.w);
}
__device__ __forceinline__ float4 f4scale(float s, float4 a) {
  return make_float4(s * a.x, s * a.y, s * a.z, s * a.w);
}
__device__ __forceinline__ float4 f4fma(float s, float4 a, float4 b) {
  return make_float4(fmaf(s, a.x, b.x), fmaf(s, a.y, b.y), fmaf(s, a.z, b.z),
                     fmaf(s, a.w, b.w));
}
__device__ __forceinline__ float f4dot(float4 a) {
  return a.x * a.x + a.y * a.y + a.z * a.z + a.w * a.w;
}

// ============================================================================
// GEMM 1&2: Out[M=262144, N=64] via K=64 with 64x64 B staged in LDS (async).
// mode 0: Out = A@Bm - Cin   (T1 = U@B - Y_H)
// mode 1: Out = Cin - LMBDA * (A@Bm)   (V = U - l*(T1@B_t))
// block = 256 threads = 8 waves; each wave: one 16-row M tile x full N=64.
// ============================================================================
__global__ void gemm_nb_kernel(const float* __restrict__ A,
                               const float* __restrict__ Bm,
                               const float* __restrict__ Cin,
                               float* __restrict__ Out, int mode) {
  __shared__ float Bs[64 * 65];  // stride 65: avoid LDS bank conflicts
  int tid = threadIdx.x;
  for (int i = tid; i < 64 * 64; i += 256) {
    int k = i >> 6, n = i & 63;
    async_b32(lds_off(Bs + k * 65 + n), Bm + i);
  }
  wait_async0();
  __syncthreads();

  int wv = tid >> 5, lane = tid & 31;
  int l = lane & 15, half = lane >> 4;
  size_t m0 = (size_t)blockIdx.x * 128 + (size_t)wv * 16;
  const float* arow = A + (m0 + l) * CC;

  v8f acc[4] = {};
  for (int kb = 0; kb < 64; kb += 4) {
    int k0 = kb + 2 * half;  // A layout: lanes 0-15 hold K=kb,kb+1; 16-31 kb+2,kb+3
    v2f a = *(const v2f*)(arow + k0);
#pragma unroll
    for (int t = 0; t < 4; ++t) {
      int bn = t * 16 + l;
      v2f b = {Bs[k0 * 65 + bn], Bs[(k0 + 1) * 65 + bn]};
      acc[t] = wmma_f32_k4(a, b, acc[t]);
    }
  }
#pragma unroll
  for (int t = 0; t < 4; ++t) {
#pragma unroll
    for (int v = 0; v < 8; ++v) {  // C/D layout: lanes>=16 hold rows M+8
      size_t idx = (m0 + v + 8 * half) * CC + t * 16 + l;
      float av = acc[t][v];
      Out[idx] = mode ? (Cin[idx] - LMBDA * av) : (av - Cin[idx]);
    }
  }
}

// ============================================================================
// GEMM 3: RU[h] = R(128x512) @ U[h](512x64) - Y_M[h].  One block per h.
// U[h] slab (512x64 f32) staged into LDS by ONE Tensor-DMA descriptor:
// tile 64x512, pad 16 DWORDs every 64 DWORDs -> LDS row stride 80 floats, so
// the +2-row half-wave offset (160 words == 32 mod 64 banks) is conflict-free.
// block = 256 threads = 8 waves; dynamic LDS = 512*80*4 = 160 KiB.
// ============================================================================
__global__ void gemm_ru_kernel(const float* __restrict__ R,
                               const float* __restrict__ U,
                               const float* __restrict__ YM,
                               float* __restrict__ RUo) {
  extern __shared__ float Us[];  // [512][80]
  int tid = threadIdx.x;
  int h = blockIdx.x;
  const float* Uh = U + (size_t)h * (WW * CC);

  if (tid == 0) {  // TDM ignores EXEC; issue once from wave 0
    // pad_iv=5: pad every 64 DWORDs; pad_amt=15: insert 16 DWORDs (64 B)
    tdm_load_2d_f32(lds_off(Us), Uh, CC, WW, CC, WW, CC, 1u, 5u, 15u);
  }
  wait_tensor0();  // no-op for waves with TENSORcnt==0
  __syncthreads();

  int wv = tid >> 5, lane = tid & 31, l = lane & 15, half = lane >> 4;
  int m0 = wv * 16;
  const float* arow = R + (size_t)(m0 + l) * WW;

  v8f acc[4] = {};
  for (int kb = 0; kb < WW; kb += 4) {
    int k0 = kb + 2 * half;
    v2f a = *(const v2f*)(arow + k0);
#pragma unroll
    for (int t = 0; t < 4; ++t) {
      int bn = t * 16 + l;
      v2f b = {Us[k0 * 80 + bn], Us[(k0 + 1) * 80 + bn]};
      acc[t] = wmma_f32_k4(a, b, acc[t]);
    }
  }
#pragma unroll
  for (int t = 0; t < 4; ++t) {
#pragma unroll
    for (int v = 0; v < 8; ++v) {
      size_t idx = (size_t)h * (WMM * CC) + (size_t)(m0 + v + 8 * half) * CC + t * 16 + l;
      RUo[idx] = acc[t][v] - YM[idx];
    }
  }
}

// ============================================================================
// GEMM 4: V[h] -= LMBDA * ( R^T(512x128) @ RU[h](128x64) ).  Batched over h.
// block = 128 threads = 4 waves; 8 blocks per h along M.
// ============================================================================
__global__ void gemm_rtru_kernel(const float* __restrict__ R,
                                 const float* __restrict__ RUi,
                                 float* __restrict__ V) {
  int tid = threadIdx.x;
  int wv = tid >> 5, lane = tid & 31, l = lane & 15, half = lane >> 4;
  int h = blockIdx.x >> 3;
  int m0 = (blockIdx.x & 7) * 64 + wv * 16;
  const float* RUh = RUi + (size_t)h * (WMM * CC);
  int m = m0 + l;

  v8f acc[4] = {};
  for (int kb = 0; kb < WMM; kb += 4) {
    int k0 = kb + 2 * half;
    v2f a = {R[(size_t)k0 * WW + m], R[(size_t)(k0 + 1) * WW + m]};  // R^T access
    if (kb + 8 < WMM)  // stream-ahead on the B operand (global_prefetch_b8)
      __builtin_prefetch(RUh + (size_t)(kb + 8) * CC + l * 4, 0, 1);
#pragma unroll
    for (int t = 0; t < 4; ++t) {
      int bn = t * 16 + l;
      v2f b = {RUh[(size_t)k0 * CC + bn], RUh[(size_t)(k0 + 1) * CC + bn]};
      acc[t] = wmma_f32_k4(a, b, acc[t]);
    }
  }
#pragma unroll
  for (int t = 0; t < 4; ++t) {
#pragma unroll
    for (int v = 0; v < 8; ++v) {
      size_t idx = (size_t)h * (WW * CC) + (size_t)(m0 + v + 8 * half) * CC + t * 16 + l;
      V[idx] -= LMBDA * acc[t][v];  // LMBDA_M == 1
    }
  }
}

// ============================================================================
// out = cd * div(scale.*z) + cv * V   (lazy projection scale applied on read)
// mode 0 (zero_ss=1): D = div + V/l ; also zeroes SS for next grad_step.
// mode 1 (zero_ss=0): U = V + l*div   (final proxg epilogue)
// ============================================================================
__global__ void div_xl_kernel(const float* __restrict__ z1,
                              const float* __restrict__ z2,
                              const float* __restrict__ Vin,
                              const float* __restrict__ scale,
                              float* __restrict__ out, float cd, float cv,
                              float* __restrict__ ss, int zero_ss) {
  int tid = threadIdx.x;
  if (zero_ss && blockIdx.x == 0) {
    for (int i = tid; i < 2 * HH; i += 256) ss[i] = 0.0f;
  }
  size_t idx = (size_t)blockIdx.x * 256 + tid;  // one float4 (4 channels) each
  int c4 = idx & 15;
  int m = (idx >> 4) & (WW - 1);
  int n = (int)(idx >> 13);
  size_t base = ((size_t)n << 15) + ((size_t)m << 6) + ((size_t)c4 << 2);

  float s0 = scale[0], s1 = scale[1];
  // d1[n] = (n<511? z1[n]:0) - (n>0? z1[n-1]:0) ; same for d2 along m
  float4 d1 = f4zero(), d2 = f4zero();
  if (n < HH - 1) d1 = ld4(z1 + base);
  if (n > 0) d1 = f4sub(d1, ld4(z1 + base - (size_t)WW * CC));
  if (m < WW - 1) d2 = ld4(z2 + base);
  if (m > 0) d2 = f4sub(d2, ld4(z2 + base - CC));

  float4 dsum = f4add(f4scale(s0, d1), f4scale(s1, d2));
  float4 r = f4fma(cv, ld4(Vin + base), f4scale(cd, dsum));
  st4(out + base, r);
}

// ============================================================================
// z <- scale.*z + 2*TAU*gradient(D); accumulate SS[s][n] = sum_{m,c} z_new^2
// block = 256 threads covers (n fixed, 16 m values, all 64 channels)
// ============================================================================
__global__ void grad_step_kernel(const float* __restrict__ D,
                                 float* __restrict__ z1, float* __restrict__ z2,
                                 const float* __restrict__ scale,
                                 float* __restrict__ ss) {
  __shared__ float r1[256];
  __shared__ float r2[256];
  int tid = threadIdx.x;
  int c4 = tid & 15, dm = tid >> 4;
  int n = blockIdx.x >> 5;
  int m = ((blockIdx.x & 31) << 4) + dm;
  size_t base = ((size_t)n << 15) + ((size_t)m << 6) + ((size_t)c4 << 2);

  float4 d0 = ld4(D + base);
  float4 gx = (n < HH - 1) ? f4sub(ld4(D + base + (size_t)WW * CC), d0) : f4zero();
  float4 gy = (m < WW - 1) ? f4sub(ld4(D + base + CC), d0) : f4zero();

  float s0 = scale[0], s1 = scale[1];
  float4 w1 = f4fma(0.2f, gx, f4scale(s0, ld4(z1 + base)));  // 2*TAU = 0.2
  float4 w2 = f4fma(0.2f, gy, f4scale(s1, ld4(z2 + base)));
  st4(z1 + base, w1);
  st4(z2 + base, w2);

  r1[tid] = f4dot(w1);
  r2[tid] = f4dot(w2);
  __syncthreads();
  for (int s = 128; s > 0; s >>= 1) {
    if (tid < s) {
      r1[tid] += r1[tid + s];
      r2[tid] += r2[tid + s];
    }
    __syncthreads();
  }
  if (tid == 0) {
    atomicAdd(&ss[n], r1[0]);
    atomicAdd(&ss[HH + n], r2[0]);
  }
}

// scale[s] = (norm>1 ? 1/norm : 1), norm = sum_n sqrt(SS[s][n])
__global__ void norm_scale_kernel(const float* __restrict__ ss,
                                  float* __restrict__ scale) {
  __shared__ float sh[512];
  int t = threadIdx.x;
  sh[t] = sqrtf(ss[t]);
  __syncthreads();
  for (int s = 256; s > 0; s >>= 1) {
    if (t < s) sh[t] += sh[t + s];
    __syncthreads();
  }
  float n0 = sh[0];
  __syncthreads();
  sh[t] = sqrtf(ss[HH + t]);
  __syncthreads();
  for (int s = 256; s > 0; s >>= 1) {
    if (t < s) sh[t] += sh[t + s];
    __syncthreads();
  }
  if (t == 0) {
    scale[0] = (n0 > 1.f) ? 1.f / n0 : 1.f;
    float n1 = sh[0];
    scale[1] = (n1 > 1.f) ? 1.f / n1 : 1.f;
  }
}

__global__ void fill4_kernel(float* __restrict__ p, float val, long long n4) {
  long long i = (long long)blockIdx.x * 256 + threadIdx.x;
  if (i < n4) st4(p + i * 4, make_float4(val, val, val, val));
}

__global__ void init_scale_kernel(float* __restrict__ scale) {
  if (threadIdx.x < 2) scale[threadIdx.x] = 1.0f;
}

// ============================================================================
extern "C" void kernel_launch(void* const* d_in, const int* in_sizes, int n_in,
                              void* d_out, int out_size, void* d_ws,
                              size_t ws_size, hipStream_t stream) {
  const float* Y_H = (const float*)d_in[0];  // (H,W,C)
  const float* Y_M = (const float*)d_in[1];  // (H,WM,C)
  const float* Bm  = (const float*)d_in[2];  // (C,C)
  const float* Bt  = (const float*)d_in[3];  // (C,C)
  const float* R   = (const float*)d_in[4];  // (WM,W)

  float* U = (float*)d_out;
  float* ws = (float*)d_ws;
  const size_t HWC = (size_t)HH * WW * CC;  // 16,777,216 floats

  float* V  = ws;                 // 64 MiB
  float* Z1 = ws + HWC;           // 64 MiB
  float* Z2 = ws + 2 * HWC;       // 64 MiB
  float* TD = ws + 3 * HWC;       // 64 MiB : T1 (grad_f phase) / D (proxg phase)
  float* RU = Z1;                 // 16 MiB, aliased (used before z is live)
  float* SS = ws + 4 * HWC;       // 2*512 floats
  float* SCALE = SS + 2 * HH;     // 2 floats

  const int EW_GRID = (int)(HWC / 4 / 256);        // 16384
  const size_t RU_LDS = (size_t)WW * 80 * sizeof(float);  // 160 KiB

  // U = 0
  fill4_kernel<<<EW_GRID, 256, 0, stream>>>(U, 0.0f, (long long)(HWC / 4));

  for (int outer = 0; outer < 5; ++outer) {
    // ---- grad_f: V = U - LMBDA * ((U@B - Y_H)@B_t + R^T(R@U - Y_M)) ----
    gemm_nb_kernel<<<2048, 256, 0, stream>>>(U, Bm, Y_H, TD, 0);
    gemm_nb_kernel<<<2048, 256, 0, stream>>>(TD, Bt, U, V, 1);
    gemm_ru_kernel<<<HH, 256, RU_LDS, stream>>>(R, U, Y_M, RU);
    gemm_rtru_kernel<<<4096, 128, 0, stream>>>(R, RU, V);

    // ---- proxg(V): TV dual ascent ----
    fill4_kernel<<<2 * EW_GRID, 256, 0, stream>>>(Z1, 1.0f,
                                                  (long long)(2 * HWC / 4));
    init_scale_kernel<<<1, 32, 0, stream>>>(SCALE);
    for (int inner = 0; inner < 5; ++inner) {
      div_xl_kernel<<<EW_GRID, 256, 0, stream>>>(Z1, Z2, V, SCALE, TD, 1.0f,
                                                 1.0f / LMBDA, SS, 1);
      grad_step_kernel<<<HH * 32, 256, 0, stream>>>(TD, Z1, Z2, SCALE, SS);
      norm_scale_kernel<<<1, 512, 0, stream>>>(SS, SCALE);
    }
    // U = V + LMBDA * div(scale.*z)
    div_xl_kernel<<<EW_GRID, 256, 0, stream>>>(Z1, Z2, V, SCALE, U, LMBDA, 1.0f,
                                               SS, 0);
  }
}